// SingleShotInhibition_25022479467194
// MI455X (gfx1250) — compile-verified
//
#include <hip/hip_runtime.h>
#include <hip/hip_bf16.h>

// SingleShotInhibition on MI455X (gfx1250):
//   out[b,i,p] = act[b,i,p] + sum_{|d|<=13} g(d) * act[b,(i+d) mod 256, p]
// Banded circular-Toeplitz matmul over channels using V_WMMA_F32_16X16X4_F32
// (full fp32). Memory-bound (206 MB @ 23.3 TB/s ~= 8.8 us floor): every global
// byte is read exactly once via async global->LDS staging (ASYNCcnt path),
// double-buffered so tile t+1 streams in while tile t runs on the matrix pipe.

typedef __attribute__((ext_vector_type(2))) float v2f;
typedef __attribute__((ext_vector_type(8))) float v8f;

#define C_CH       256
#define HW_PIX     3136          // 56*56
#define BATCH      32
#define HALF_SCOPE 13            // (27-1)/2
#define PIX_TILE   16
#define KSTEPS     12            // K window = 48 columns: j in [i0-16, i0+32)
#define NT         4             // pixel tiles per block (double buffered)
#define TILE_FLOATS (C_CH * PIX_TILE)   // 4096 floats = 16 KB per buffer

// Issue one 16-pixel-tile stage (256 channels x 16 pixels = 16 KB) as
// 4 rounds of GLOBAL_LOAD_ASYNC_TO_LDS_B128 per thread (16 B per lane).
// Completion tracked by ASYNCcnt; each wave issues exactly 4 async ops.
__device__ __forceinline__ void stage_async(const float* bufBase,   // LDS (generic ptr)
                                            const float* gBase,     // act + b*C*HW
                                            int p0t, int tid)
{
#pragma unroll
    for (int r = 0; r < 4; ++r) {
        const int f = r * 256 + tid;     // 0..1023
        const int c = f >> 2;            // channel row 0..255
        const int g = f & 3;             // 16B group within the 64B row
        // Low 32 bits of a generic LDS pointer are the LDS byte address
        const unsigned ldsOff = (unsigned)(size_t)(bufBase + c * PIX_TILE + g * 4);
        const unsigned gOff   = (unsigned)(((unsigned)c * HW_PIX + (unsigned)p0t + g * 4) * 4u);
        asm volatile("global_load_async_to_lds_b128 %0, %1, %2"
                     :: "v"(ldsOff), "v"(gOff), "s"(gBase)
                     : "memory");
    }
}

__global__ __launch_bounds__(256)
void inhib_wmma_kernel(const float* __restrict__ act,
                       const float* __restrict__ filt,   // 27 taps, center already 0
                       float* __restrict__ out)
{
    __shared__ float sh[2 * TILE_FLOATS];   // 32 KB, double buffered

    const int blkPerBatch = HW_PIX / (PIX_TILE * NT);   // 49
    const int blk = blockIdx.x;
    const int b   = blk / blkPerBatch;
    const int p0  = (blk % blkPerBatch) * (PIX_TILE * NT);

    const int tid  = threadIdx.x;
    const int lane = tid & 31;
    const int wave = tid >> 5;

    const float* gBase = act + (size_t)b * C_CH * HW_PIX;

    // ---- kick off stage of tile 0 while we set up A fragments ----
    stage_async(sh, gBase, p0, tid);

    // ---- per-lane A fragments (Toeplitz band slice); identical for all tiles ----
    // 16x4 f32 A layout: lanes 0-15 hold M=lane with K=0 (v0), K=1 (v1);
    //                    lanes 16-31 hold M=lane-16 with K=2 (v0), K=3 (v1).
    // A[M,K] = tpl[i0+M, i0-16+4*kk+K] = g(4*kk + K - 16 - M), origin-independent.
    const int M  = lane & 15;
    const int kb = (lane < 16) ? 0 : 2;

    v2f afrag[KSTEPS];
#pragma unroll
    for (int kk = 0; kk < KSTEPS; ++kk) {
        const int d0 = 4 * kk + kb - 16 - M;
        const int d1 = d0 + 1;
        float a0 = 0.0f, a1 = 0.0f;
        if (d0 >= -HALF_SCOPE && d0 <= HALF_SCOPE) a0 = filt[d0 + HALF_SCOPE];
        if (d1 >= -HALF_SCOPE && d1 <= HALF_SCOPE) a1 = filt[d1 + HALF_SCOPE];
        afrag[kk].x = a0;
        afrag[kk].y = a1;
    }

    const int col    = lane & 15;             // pixel within tile (N index)
    const int rowAdd = (lane < 16) ? 0 : 8;   // C/D layout: vgpr r -> M = r (+8 hi half)

    for (int t = 0; t < NT; ++t) {
        if (t + 1 < NT) {
            // prefetch next tile into the other buffer (read hazard cleared by
            // the trailing barrier of iteration t-1), then wait only for the
            // *current* tile's 4 async ops (in-order completion => <=4 is exact)
            stage_async(sh + ((t + 1) & 1) * TILE_FLOATS, gBase,
                        p0 + (t + 1) * PIX_TILE, tid);
            asm volatile("s_wait_asynccnt 0x4" ::: "memory");
        } else {
            asm volatile("s_wait_asynccnt 0x0" ::: "memory");
        }
        __syncthreads();   // all waves' async writes for tile t are in LDS

        const float* buf = sh + (t & 1) * TILE_FLOATS;
        const int p0t = p0 + t * PIX_TILE;

        // ---- each wave computes two 16x16 (channels x pixels) output tiles ----
#pragma unroll
        for (int tt = 0; tt < 2; ++tt) {
            const int i0 = wave * 32 + tt * 16;    // tile channel origin

            v8f acc = {};
#pragma unroll
            for (int kk = 0; kk < KSTEPS; ++kk) {
                // B (4x16, K x N) layout mirrors A: lanes 0-15 -> K=0 (v0), K=1 (v1);
                //                                   lanes 16-31 -> K=2 (v0), K=3 (v1).
                const int jb = i0 - 16 + 4 * kk + kb;   // circular band window
                v2f bfrag;
                bfrag.x = buf[((jb    ) & (C_CH - 1)) * PIX_TILE + col];
                bfrag.y = buf[((jb + 1) & (C_CH - 1)) * PIX_TILE + col];

                acc = __builtin_amdgcn_wmma_f32_16x16x4_f32(
                    /*neg_a=*/false, afrag[kk],
                    /*neg_b=*/false, bfrag,
                    /*c_mod=*/(short)0, acc,
                    /*reuse_a=*/false, /*reuse_b=*/false);
            }

            // identity add (out = act + conv) + coalesced half-wave row stores
#pragma unroll
            for (int r = 0; r < 8; ++r) {
                const int row = i0 + r + rowAdd;
                const float v = acc[r] + buf[row * PIX_TILE + col];
                out[((size_t)b * C_CH + row) * HW_PIX + p0t + col] = v;
            }
        }

        __syncthreads();   // everyone done reading buf before it is restaged
    }
}

extern "C" void kernel_launch(void* const* d_in, const int* in_sizes, int n_in,
                              void* d_out, int out_size, void* d_ws, size_t ws_size,
                              hipStream_t stream)
{
    const float* act  = (const float*)d_in[0];   // [32, 256, 56, 56] fp32
    const float* filt = (const float*)d_in[1];   // [27] fp32
    float* out = (float*)d_out;                  // [32, 256, 56, 56] fp32

    const int blocks = BATCH * (HW_PIX / (PIX_TILE * NT));   // 32 * 49 = 1568
    inhib_wmma_kernel<<<blocks, 256, 0, stream>>>(act, filt, out);
}